// MultiScaleTrendDirectionLoss_910533067633
// MI455X (gfx1250) — compile-verified
//
#include <hip/hip_runtime.h>
#include <hip/hip_bf16.h>

// ---------------------------------------------------------------------------
// MultiScaleTrendDirectionLoss for MI455X (gfx1250, wave32)
// Bandwidth-bound streaming scan:  async global->LDS double buffering,
// exact blocked linear-recurrence scan via cross-lane affine Hillis-Steele.
// ---------------------------------------------------------------------------

#define SLEN  8192
#define TILE  1024
#define NTILE (SLEN / TILE)          // 8 tiles per row
#define LPT   (TILE / 32)            // 32 elements per lane per tile
// LDS padding: 4 floats (16B) per 32-float group -> async B128 stays 16B aligned
#define PADF(i) ((i) + ((((i) >> 5)) << 2))
#define PTILE (PADF(TILE) + 8)       // 1160 floats per buffer (slack for guarded overread)

#if defined(__HIP_DEVICE_COMPILE__) && __has_builtin(__builtin_amdgcn_global_load_async_to_lds_b128)
#define USE_ASYNC 1
#else
#define USE_ASYNC 0
#endif

#if USE_ASYNC
// int4 vector pointee types with explicit address spaces, per the builtin's
// signature: (v4i __device__*, v4i __shared__*, imm offset, imm cpol)
typedef int v4i __attribute__((vector_size(16)));
typedef v4i __attribute__((address_space(1)))* v4i_gptr;
typedef v4i __attribute__((address_space(3)))* v4i_lptr;
#define WAIT_TILE() asm volatile("s_wait_asynccnt 0x10" ::: "memory")
#define WAIT_ALL()  asm volatile("s_wait_asynccnt 0x0"  ::: "memory")
#else
#define WAIT_TILE() do {} while (0)
#define WAIT_ALL()  do {} while (0)
#endif

namespace msl {
constexpr double ALPHA_D[3] = {0.1, 0.3, 0.6};
constexpr float  WGT[3]     = {0.5f, 0.3f, 0.2f};

constexpr float fpow(double b, int n) {
  double r = 1.0;
  for (int i = 0; i < n; ++i) r *= b;
  return (float)r;
}

struct Tab {
  float a[3];      // alpha
  float b[3];      // 1 - alpha
  float m[3][5];   // beta^(32<<k) : affine-scan multipliers for strides 1,2,4,8,16
};
constexpr Tab make_tab() {
  Tab t{};
  for (int i = 0; i < 3; ++i) {
    t.a[i] = (float)ALPHA_D[i];
    t.b[i] = (float)(1.0 - ALPHA_D[i]);
    for (int k = 0; k < 5; ++k) t.m[i][k] = fpow(1.0 - ALPHA_D[i], LPT << k);
  }
  return t;
}
constexpr Tab TAB = make_tab();
} // namespace msl

__device__ __forceinline__ float huber_term(float pd, float td) {
  float dir  = pd * td;
  float step = fabsf(pd - td);
  float mrg  = fmaxf(1.0f - dir, 0.0f);
  float quad = 0.5f * mrg * mrg;   // HUBER_MARGIN == 1
  return (dir < 0.0f) ? step : quad;
}

// Stage one TILE of one signal into (padded) LDS.
__device__ __forceinline__ void issue_tile(const float* __restrict__ g,
                                           float* lds, int lane) {
#if USE_ASYNC
#pragma unroll
  for (int q = 0; q < TILE / (4 * 32); ++q) {      // 8 async b128 ops / signal / tile
    int c = q * 32 + lane;                          // 16B chunk id, coalesced per op
    v4i_gptr gp = (v4i_gptr)(g + c * 4);
    v4i_lptr lp = (v4i_lptr)(lds + c * 4 + ((c >> 3) << 2));
    __builtin_amdgcn_global_load_async_to_lds_b128(gp, lp, 0, 0);
  }
#else
#pragma unroll
  for (int q = 0; q < LPT; ++q) {
    int i = q * 32 + lane;
    lds[PADF(i)] = g[i];
  }
#endif
}

// One wave per row. Exact blocked scan of 6 recurrences (3 alphas x 2 signals).
__global__ __launch_bounds__(32) void msl_row_kernel(const float* __restrict__ pred,
                                                     const float* __restrict__ targ,
                                                     float* __restrict__ partial) {
  __shared__ float sp[2][PTILE];
  __shared__ float st[2][PTILE];
  using msl::TAB;

  const int row  = blockIdx.x;
  const int lane = threadIdx.x;
  const float* gp = pred + (size_t)row * SLEN;
  const float* gt = targ + (size_t)row * SLEN;

  float cp[3] = {0.f, 0.f, 0.f}, ct[3] = {0.f, 0.f, 0.f};   // row carries (uniform)
  float acc[3] = {0.f, 0.f, 0.f};

  issue_tile(gp, sp[0], lane);
  issue_tile(gt, st[0], lane);

  for (int tile = 0; tile < NTILE; ++tile) {
    const int buf = tile & 1;
    if (tile + 1 < NTILE) {
      issue_tile(gp + (tile + 1) * TILE, sp[buf ^ 1], lane);
      issue_tile(gt + (tile + 1) * TILE, st[buf ^ 1], lane);
      WAIT_TILE();                      // tile k landed; tile k+1 still in flight
    } else {
      WAIT_ALL();
    }
    const float* p = sp[buf];
    const float* t = st[buf];
    const int base = lane * LPT;

    float ocp[3], oct[3];
#pragma unroll
    for (int a = 0; a < 3; ++a) { ocp[a] = cp[a]; oct[a] = ct[a]; }

    // Cross-tile boundary loss term: t = tile*TILE - 1 (needs x[tile*TILE]).
    if (tile > 0 && lane == 0) {
      float xnp = p[0], xnt = t[0];
#pragma unroll
      for (int a = 0; a < 3; ++a)
        acc[a] += huber_term(TAB.a[a] * (xnp - ocp[a]), TAB.a[a] * (xnt - oct[a]));
    }

    // ---- pass 1: zero-carry local scans (32 elements / lane) ----
    float Ap[3] = {0.f, 0.f, 0.f}, At[3] = {0.f, 0.f, 0.f};
#pragma unroll 8
    for (int j = 0; j < LPT; ++j) {
      float xp = p[PADF(base + j)];
      float xt = t[PADF(base + j)];
      if (tile == 0 && lane == 0 && j == 0) {   // reference: e[0] = x[0]
#pragma unroll
        for (int a = 0; a < 3; ++a) { Ap[a] = xp; At[a] = xt; }
      } else {
#pragma unroll
        for (int a = 0; a < 3; ++a) {
          Ap[a] = fmaf(TAB.b[a], Ap[a], TAB.a[a] * xp);
          At[a] = fmaf(TAB.b[a], At[a], TAB.a[a] * xt);
        }
      }
    }

    // Fold previous-tile carry into lane 0 before the cross-lane scan.
    if (tile > 0 && lane == 0) {
#pragma unroll
      for (int a = 0; a < 3; ++a) {
        Ap[a] = fmaf(TAB.m[a][0], ocp[a], Ap[a]);
        At[a] = fmaf(TAB.m[a][0], oct[a], At[a]);
      }
    }

    // ---- exact affine Hillis-Steele across 32 lanes:  y_l = A_l + M*y_{l-1} ----
#pragma unroll
    for (int a = 0; a < 3; ++a) {
#pragma unroll
      for (int k = 0; k < 5; ++k) {
        const int s = 1 << k;
        float upP = __shfl_up(Ap[a], s, 32);
        float upT = __shfl_up(At[a], s, 32);
        if (lane >= s) {
          Ap[a] = fmaf(TAB.m[a][k], upP, Ap[a]);
          At[a] = fmaf(TAB.m[a][k], upT, At[a]);
        }
      }
    }

    // Per-lane carry-in (previous lane's last EMA value) + new row carries.
    float ep[3], et[3];
#pragma unroll
    for (int a = 0; a < 3; ++a) {
      float upP = __shfl_up(Ap[a], 1, 32);
      float upT = __shfl_up(At[a], 1, 32);
      ep[a] = (lane == 0) ? ocp[a] : upP;
      et[a] = (lane == 0) ? oct[a] : upT;
      cp[a] = __shfl(Ap[a], 31, 32);
      ct[a] = __shfl(At[a], 31, 32);
    }

    // ---- pass 2: rescan with exact carries, accumulate loss on the fly ----
    float xp = p[PADF(base)];
    float xt = t[PADF(base)];
#pragma unroll 8
    for (int j = 0; j < LPT; ++j) {
      if (tile == 0 && lane == 0 && j == 0) {
#pragma unroll
        for (int a = 0; a < 3; ++a) { ep[a] = xp; et[a] = xt; }
      } else {
#pragma unroll
        for (int a = 0; a < 3; ++a) {
          ep[a] = fmaf(TAB.b[a], ep[a], TAB.a[a] * xp);
          et[a] = fmaf(TAB.b[a], et[a], TAB.a[a] * xt);
        }
      }
      float xnp = p[PADF(base + j + 1)];      // x[t+1]; lane31/j31 overread is guarded
      float xnt = t[PADF(base + j + 1)];
      if (!(lane == 31 && j == LPT - 1)) {    // cross-tile term handled at next tile
#pragma unroll
        for (int a = 0; a < 3; ++a)
          acc[a] += huber_term(TAB.a[a] * (xnp - ep[a]), TAB.a[a] * (xnt - et[a]));
      }
      xp = xnp; xt = xnt;
    }
  }

  float total = msl::WGT[0] * acc[0] + msl::WGT[1] * acc[1] + msl::WGT[2] * acc[2];
#pragma unroll
  for (int s = 16; s > 0; s >>= 1) total += __shfl_xor(total, s, 32);
  if (lane == 0) partial[row] = total;
}

// Deterministic final reduction (single block, fixed tree; no float atomics).
__global__ __launch_bounds__(256) void msl_reduce_kernel(const float* __restrict__ partial,
                                                         float* __restrict__ out, int rows) {
  __shared__ float sm[256];
  float s = 0.f;
  for (int i = threadIdx.x; i < rows; i += 256) s += partial[i];
  sm[threadIdx.x] = s;
  __syncthreads();
  for (int st = 128; st > 0; st >>= 1) {
    if ((int)threadIdx.x < st) sm[threadIdx.x] += sm[threadIdx.x + st];
    __syncthreads();
  }
  if (threadIdx.x == 0)
    out[0] = sm[0] * (1.0f / ((float)rows * (float)(SLEN - 1)));
}

extern "C" void kernel_launch(void* const* d_in, const int* in_sizes, int n_in,
                              void* d_out, int out_size, void* d_ws, size_t ws_size,
                              hipStream_t stream) {
  const float* pred = (const float*)d_in[0];
  const float* targ = (const float*)d_in[1];
  float* ws  = (float*)d_ws;
  float* out = (float*)d_out;
  const int rows = in_sizes[0] / SLEN;   // 2048

  msl_row_kernel<<<dim3(rows), dim3(32), 0, stream>>>(pred, targ, ws);
  msl_reduce_kernel<<<dim3(1), dim3(256), 0, stream>>>(ws, out, rows);
}